// HPOLoss_14439680049772
// MI455X (gfx1250) — compile-verified
//
#include <hip/hip_runtime.h>

// Problem constants (from the reference)
#define WW 26
#define HH 26
#define DDEPTH 5
#define GG 3380                    // W*H*DD cells per batch
#define CH_STRIDE 3380             // floats between channels
#define BATCH_STRIDE 216320        // 64 channels * 3380 (== 320*26*26)
#define NGROUP 845                 // GG / 4 (float4 groups of cells)
#define CHUNKS 4                   // ceil(NGROUP / 256)
#define L2E 1.4426950408889634f    // log2(e)

typedef __attribute__((ext_vector_type(4))) float vf4;
typedef __attribute__((ext_vector_type(2))) float v2f;
typedef __attribute__((ext_vector_type(8))) float v8f;

__device__ __forceinline__ float fast_sigmoid(float x) {
    // 1/(1+e^-x) via v_exp_f32 + v_rcp_f32 (1-ulp class)
    return __builtin_amdgcn_rcpf(1.0f + __builtin_amdgcn_exp2f(-x * L2E));
}

// ---------------------------------------------------------------------------
// Stage 1 (hot): confidence loss over every (batch, cell). Each thread owns 4
// consecutive cells -> 64 b128 NT loads streaming all 64 channels. Lean inner
// loop: per joint/cell = 3 sub + 6 fma + sqrt + exp + fma + cmp/select.
// The GT-cell coordinate loss is NOT here (hoisted to stage 1b).
// ---------------------------------------------------------------------------
__global__ __launch_bounds__(256)
void hpo_conf_kernel(const float* __restrict__ pred,
                     const float* __restrict__ uvd_gt,
                     float* __restrict__ partials)
{
    __shared__ float gts[63];      // gt pre-scaled by per-axis pixel scale
    __shared__ float red[256];
    __shared__ int   s_lin;

    const int b   = blockIdx.y;
    const int tid = threadIdx.x;

    if (tid < 63) {
        const float v  = uvd_gt[b * 63 + tid];
        const int axis = tid % 3;
        const float sc = (axis == 0) ? 1920.0f : (axis == 1 ? 1080.0f : 1000.0f);
        gts[tid] = v * sc;
    }
    if (tid == 0) {
        const float g0 = uvd_gt[b * 63 + 0];
        const float g1 = uvd_gt[b * 63 + 1];
        const float g2 = uvd_gt[b * 63 + 2];
        int gi = (int)(g0 * 26.0f); gi = gi < 0 ? 0 : (gi > 25 ? 25 : gi);
        int gj = (int)(g1 * 26.0f); gj = gj < 0 ? 0 : (gj > 25 ? 25 : gj);
        int gk = (int)(g2 * 5.0f);  gk = gk < 0 ? 0 : (gk > 4  ? 4  : gk);
        s_lin = 130 * gi + 5 * gj + gk;
    }
    __syncthreads();
    const int lin = s_lin;

    const int grp = blockIdx.x * 256 + tid;   // float4 group of 4 cells
    float tsum = 0.0f;

    if (grp < NGROUP) {
        const int g0 = grp * 4;

        // per-sub-cell axis base: coord * (scale/dim)
        float au[4], av[4], az[4];
        #pragma unroll
        for (int c = 0; c < 4; ++c) {
            const int g   = g0 + c;
            const int iu  = g / 130;
            const int rem = g - iu * 130;
            const int iv  = rem / 5;
            const int iz  = rem - iv * 5;
            au[c] = (float)iu * (1920.0f / 26.0f);
            av[c] = (float)iv * (1080.0f / 26.0f);
            az[c] = (float)iz * (1000.0f / 5.0f);
        }

        const float* pb  = pred + (size_t)b * BATCH_STRIDE + g0;
        const float  K1   = 2.0f * L2E;              // exp(2-2d/75) as exp2
        const float  K2   = (2.0f / 75.0f) * L2E;
        const float  inv0 = 1.0f / (expf(2.0f) - 1.0f);

        float conf[4] = {0.0f, 0.0f, 0.0f, 0.0f};

        #pragma unroll
        for (int j = 0; j < 21; ++j) {
            vf4 x = __builtin_nontemporal_load((const vf4*)(pb + (3 * j + 0) * CH_STRIDE));
            vf4 y = __builtin_nontemporal_load((const vf4*)(pb + (3 * j + 1) * CH_STRIDE));
            vf4 w = __builtin_nontemporal_load((const vf4*)(pb + (3 * j + 2) * CH_STRIDE));
            if (j == 0) {  // HAND_ROOT gets sigmoid
                #pragma unroll
                for (int c = 0; c < 4; ++c) {
                    x[c] = fast_sigmoid(x[c]);
                    y[c] = fast_sigmoid(y[c]);
                    w[c] = fast_sigmoid(w[c]);
                }
            }
            const float gu = gts[3 * j + 0];
            const float gv = gts[3 * j + 1];
            const float gz = gts[3 * j + 2];
            #pragma unroll
            for (int c = 0; c < 4; ++c) {
                const float du = fmaf(x[c], 1920.0f / 26.0f, au[c] - gu);
                const float dv = fmaf(y[c], 1080.0f / 26.0f, av[c] - gv);
                const float dz = fmaf(w[c], 1000.0f / 5.0f,  az[c] - gz);
                float d2 = fmaf(du, du, 1e-5f);
                d2 = fmaf(dv, dv, d2);
                d2 = fmaf(dz, dz, d2);
                const float dist = __builtin_amdgcn_sqrtf(d2);
                const float e    = __builtin_amdgcn_exp2f(fmaf(dist, -K2, K1));
                const float cadd = fmaf(e, inv0, -inv0);     // (e^{..}-1)/(e^2-1)
                conf[c] += (dist < 75.0f) ? cadd : 0.0f;     // continuous at 75
            }
        }

        const vf4 pc = __builtin_nontemporal_load((const vf4*)(pb + 63 * CH_STRIDE));
        #pragma unroll
        for (int c = 0; c < 4; ++c) {
            const float mean_conf = conf[c] * (1.0f / 21.0f);
            const float pconf     = fast_sigmoid(pc[c]);
            const bool  is_lin    = (g0 + c == lin);
            const float cm2 = is_lin ? 5.0f : ((mean_conf > 0.6f) ? 0.0f : 0.1f);
            const float tc  = is_lin ? mean_conf : 0.0f;
            const float cd  = pconf - tc;
            tsum += 0.5f * cm2 * cd * cd;
        }
    }

    red[tid] = tsum;
    __syncthreads();
    #pragma unroll
    for (int s = 128; s > 0; s >>= 1) {
        if (tid < s) red[tid] += red[tid + s];
        __syncthreads();
    }
    if (tid == 0) partials[blockIdx.y * CHUNKS + blockIdx.x] = red[0];
}

// ---------------------------------------------------------------------------
// Stage 1b (cold): per-batch coordinate loss at the GT root cell only.
// 256 blocks x 64 threads; thread c < 63 handles one (joint, axis) channel.
// ---------------------------------------------------------------------------
__global__ __launch_bounds__(64)
void hpo_uvd_kernel(const float* __restrict__ pred,
                    const float* __restrict__ uvd_gt,
                    float* __restrict__ partials, int pofs)
{
    __shared__ float red[64];
    const int b = blockIdx.x;
    const int c = threadIdx.x;

    const float g0 = uvd_gt[b * 63 + 0];
    const float g1 = uvd_gt[b * 63 + 1];
    const float g2 = uvd_gt[b * 63 + 2];
    int gi = (int)(g0 * 26.0f); gi = gi < 0 ? 0 : (gi > 25 ? 25 : gi);
    int gj = (int)(g1 * 26.0f); gj = gj < 0 ? 0 : (gj > 25 ? 25 : gj);
    int gk = (int)(g2 * 5.0f);  gk = gk < 0 ? 0 : (gk > 4  ? 4  : gk);
    const int lin = 130 * gi + 5 * gj + gk;

    float val = 0.0f;
    if (c < 63) {
        float x = pred[(size_t)b * BATCH_STRIDE + c * CH_STRIDE + lin];
        if (c < 3) x = fast_sigmoid(x);             // root joint
        const int   axis = c % 3;
        const float dim  = (axis == 0) ? 26.0f : (axis == 1 ? 26.0f : 5.0f);
        const float gidx = (axis == 0) ? (float)gi : (axis == 1 ? (float)gj : (float)gk);
        const float tval = uvd_gt[b * 63 + c] * dim - gidx;
        const float d    = x - tval;
        val = 0.5f * d * d;
    }
    red[c] = val;
    __syncthreads();
    #pragma unroll
    for (int s = 32; s > 0; s >>= 1) {
        if (c < s) red[c] += red[c + s];
        __syncthreads();
    }
    if (c == 0) partials[pofs + b] = red[0];
}

// ---------------------------------------------------------------------------
// Stage 2: single wave32. Deterministic strided accumulation, then cross-lane
// sum via V_WMMA_F32_16X16X4_F32 against an all-ones B matrix:
// A VGPR0 = acc (K=0 lanes 0-15, K=2 lanes 16-31), A VGPR1 = 0, so
// D[m,n] = acc_m + acc_{m+16}; low-lane + high-lane D-register sums give the
// full 32-lane total.
// ---------------------------------------------------------------------------
__global__ __launch_bounds__(32)
void hpo_final_reduce(const float* __restrict__ partials,
                      float* __restrict__ out, int n)
{
    const int lane = threadIdx.x;
    float acc = 0.0f;
    for (int i = lane; i < n; i += 32) acc += partials[i];

    v2f a;  a[0] = acc;  a[1] = 0.0f;
    v2f bm; bm[0] = 1.0f; bm[1] = 1.0f;
    v8f c = {};
    c = __builtin_amdgcn_wmma_f32_16x16x4_f32(
            /*neg_a=*/false, a, /*neg_b=*/false, bm,
            /*c_mod=*/(short)0, c, /*reuse_a=*/false, /*reuse_b=*/false);

    float s = c[0] + c[1] + c[2] + c[3] + c[4] + c[5] + c[6] + c[7];
    const float lo = __shfl(s, 0, 32);
    const float hi = __shfl(s, 16, 32);
    if (lane == 0) out[0] = lo + hi;
}

extern "C" void kernel_launch(void* const* d_in, const int* in_sizes, int n_in,
                              void* d_out, int out_size, void* d_ws, size_t ws_size,
                              hipStream_t stream)
{
    const float* pred = (const float*)d_in[0];
    const float* gt   = (const float*)d_in[1];
    // d_in[2] = train_out (0) -> single scalar output

    const int B = in_sizes[1] / 63;               // 256
    float* partials = (float*)d_ws;               // CHUNKS*B + B floats (~5 KB)

    dim3 grid(CHUNKS, B);
    hpo_conf_kernel<<<grid, 256, 0, stream>>>(pred, gt, partials);
    hpo_uvd_kernel<<<B, 64, 0, stream>>>(pred, gt, partials, CHUNKS * B);
    hpo_final_reduce<<<1, 32, 0, stream>>>(partials, (float*)d_out, CHUNKS * B + B);
}